// CondGCN_13804024889951
// MI455X (gfx1250) — compile-verified
//
#include <hip/hip_runtime.h>
#include <hip/hip_bf16.h>
#include <cstddef>

// ---------------------------------------------------------------------------
// CondGCN on gfx1250 (MI455X)
//
// Rewrite: relu(x[gather] @ W.T + b) == relu(x @ W.T + b)[gather], so all
// linear layers run once per *node* (450K rows total instead of 3M edge rows).
// GEMMs use V_WMMA_F32_16X16X4_F32 (full f32, matches reference precision);
// the edge phase is pure gather + global_atomic_add_f32 scatter, which is the
// L2-bandwidth bottleneck (h buffers ~64MB + agg 25.6MB fit in 192MB L2).
//
// v2: prefetch all 16 A fragments per tile before the WMMA chain so the
// compiler emits incremental s_wait_loadcnt waits instead of a full stall
// per k-step (occupancy is ~1 wave/SIMD because B lives in ~384 VGPRs, so
// in-wave pipelining is the only latency hiding available).
// ---------------------------------------------------------------------------

#define NX_   100000
#define NC_   50000
#define DH_   64

typedef __attribute__((ext_vector_type(2))) float v2f;
typedef __attribute__((ext_vector_type(8))) float v8f;

// ---------------------------------------------------------------------------
// Wave-tile GEMM: out_m = act(in[N x 64] @ W_m[64 x 64]^T + b_m), m < NMAT.
// One wave owns a 16-row tile; 4 col-tiles of 16; K=64 -> 16 wmma steps.
//
// f32 WMMA 16x16x4 operand layouts (ISA 7.12.2):
//   A (16x4): lane L<16 holds row L, K = {4kk, 4kk+1}; lane L>=16 holds
//             row L-16, K = {4kk+2, 4kk+3}  -> one float2 load per lane.
//   B (4x16): lane L holds col N = L%16; lanes<16: K = {4kk,4kk+1},
//             lanes>=16: K = {4kk+2,4kk+3}. B[k][n] = W[n][k] -> float2 of W.
//   D (16x16): VGPR v, lane L -> row = v + 8*(L/16), col = L%16.
//
// Weights are pre-swizzled into LDS in exactly the per-lane B order; the
// compiler hoists them into VGPRs across the tile loop (conflict-free
// ds_load_b64 once per weight tile).
// ---------------------------------------------------------------------------
template<int NMAT, bool RELU>
__global__ __launch_bounds__(128) void gemm64_multi(
    const float* __restrict__ in, int ntiles,
    const float* __restrict__ W0, const float* __restrict__ b0, float* __restrict__ o0,
    const float* __restrict__ W1, const float* __restrict__ b1, float* __restrict__ o1,
    const float* __restrict__ W2, const float* __restrict__ b2, float* __restrict__ o2)
{
    __shared__ float bsh[NMAT * 4096];   // NMAT * 16KB, swizzled B layout

    // ---- fill LDS with swizzled weights (compile-time m -> no ptr array) ----
    #pragma unroll
    for (int m = 0; m < NMAT; ++m) {
        const float* __restrict__ W = (m == 0) ? W0 : ((m == 1) ? W1 : W2);
        for (int r = threadIdx.x; r < 4096; r += blockDim.x) {
            int n = r >> 6;          // output col  (row of W)
            int k = r & 63;          // reduction k (col of W)
            float val = W[n * 64 + k];
            int kk = k >> 2;                    // wmma k-step
            int hi = (k >> 1) & 1;              // lane half
            int lo = k & 1;                     // which of the 2 VGPRs
            int j  = n >> 4;                    // col-tile
            int ln = (n & 15) + (hi << 4);      // lane within wave
            bsh[((((m * 16 + kk) * 4 + j) * 32) + ln) * 2 + lo] = val;
        }
    }
    __syncthreads();

    const int w     = threadIdx.x >> 5;
    const int lane  = threadIdx.x & 31;
    const int lhalf = lane >> 4;     // 0 / 1
    const int llo   = lane & 15;
    const float2* __restrict__ bsh2 = reinterpret_cast<const float2*>(bsh);

    for (int t = blockIdx.x * 4 + w; t < ntiles; t += gridDim.x * 4) {
        v8f zero = {};
        v8f acc[NMAT][4];
        #pragma unroll
        for (int m = 0; m < NMAT; ++m)
            #pragma unroll
            for (int j = 0; j < 4; ++j)
                acc[m][j] = zero;

        // ---- prefetch the whole A tile slice for this lane (16 float2) ----
        // 16 back-to-back global_load_b64; the wmma loop below then drains
        // them with incremental s_wait_loadcnt instead of one stall per step.
        const float* __restrict__ arow =
            in + (size_t)(t * 16 + llo) * 64 + (lhalf << 1);
        v2f a[16];
        #pragma unroll
        for (int kk = 0; kk < 16; ++kk) {
            float2 af = *reinterpret_cast<const float2*>(arow + kk * 4);
            a[kk].x = af.x;
            a[kk].y = af.y;
        }

        #pragma unroll
        for (int kk = 0; kk < 16; ++kk) {
            #pragma unroll
            for (int m = 0; m < NMAT; ++m) {
                #pragma unroll
                for (int j = 0; j < 4; ++j) {
                    float2 bf = bsh2[(((m * 16 + kk) * 4 + j) << 5) + lane];
                    v2f B; B.x = bf.x; B.y = bf.y;
                    acc[m][j] = __builtin_amdgcn_wmma_f32_16x16x4_f32(
                        /*neg_a=*/false, a[kk], /*neg_b=*/false, B,
                        /*c_mod=*/(short)0, acc[m][j],
                        /*reuse_a=*/false, /*reuse_b=*/false);
                }
            }
        }

        // ---- epilogue: bias (+ relu), store D tiles ----
        #pragma unroll
        for (int m = 0; m < NMAT; ++m) {
            const float* __restrict__ bm = (m == 0) ? b0 : ((m == 1) ? b1 : b2);
            float* __restrict__       om = (m == 0) ? o0 : ((m == 1) ? o1 : o2);
            #pragma unroll
            for (int j = 0; j < 4; ++j) {
                int col = (j << 4) + llo;
                float bias = bm[col];
                float* __restrict__ op =
                    om + (size_t)(t * 16 + (lhalf << 3)) * 64 + col;
                #pragma unroll
                for (int v = 0; v < 8; ++v) {
                    float o = acc[m][j][v] + bias;
                    if (RELU) o = fmaxf(o, 0.0f);
                    op[v * 64] = o;   // row = t*16 + 8*lhalf + v
                }
            }
        }
    }
}

// ---------------------------------------------------------------------------
// Edge scatter: agg[dst] += h[src] per edge, 64 f32 per edge.
// 16 threads/edge, float4 gather + 4x global_atomic_add_f32 scatter.
// ---------------------------------------------------------------------------
__global__ __launch_bounds__(256) void scatter_add64(
    const float* __restrict__ h, const int* __restrict__ gidx,
    const int* __restrict__ sidx, float* __restrict__ agg, int E)
{
    int tid = blockIdx.x * blockDim.x + threadIdx.x;
    int e = tid >> 4;
    if (e >= E) return;
    int q = (tid & 15) << 2;
    int src = gidx[e];
    int dst = sidx[e];
    const float4 v = *reinterpret_cast<const float4*>(h + (size_t)src * 64 + q);
    float* p = agg + (size_t)dst * 64 + q;
    atomicAdd(p + 0, v.x);
    atomicAdd(p + 1, v.y);
    atomicAdd(p + 2, v.z);
    atomicAdd(p + 3, v.w);
}

// ---------------------------------------------------------------------------
// Host launch. Input order (setup_inputs):
//  0:x 1:c 2:e_fwd_src 3:e_fwd_dst 4:e_bwd_src 5:e_bwd_dst
//  6:e_cx_src 7:e_cx_dst 8:Wx 9:bx 10:Wc 11:bc 12:Wff 13:bff
//  14:Wbb 15:bbb 16:Wcx 17:bcx 18:Wp 19:bp
// d_out = [out_x (NX*64) | out_c (NC*64)] f32.
// ---------------------------------------------------------------------------
extern "C" void kernel_launch(void* const* d_in, const int* in_sizes, int n_in,
                              void* d_out, int out_size, void* d_ws, size_t ws_size,
                              hipStream_t stream) {
    const float* x  = (const float*)d_in[0];
    const float* c  = (const float*)d_in[1];
    const int* e_fwd_src = (const int*)d_in[2];
    const int* e_fwd_dst = (const int*)d_in[3];
    const int* e_bwd_src = (const int*)d_in[4];
    const int* e_bwd_dst = (const int*)d_in[5];
    const int* e_cx_src  = (const int*)d_in[6];
    const int* e_cx_dst  = (const int*)d_in[7];
    const float* Wx  = (const float*)d_in[8];
    const float* bx  = (const float*)d_in[9];
    const float* Wc  = (const float*)d_in[10];
    const float* bc  = (const float*)d_in[11];
    const float* Wff = (const float*)d_in[12];
    const float* bff = (const float*)d_in[13];
    const float* Wbb = (const float*)d_in[14];
    const float* bbb = (const float*)d_in[15];
    const float* Wcx = (const float*)d_in[16];
    const float* bcx = (const float*)d_in[17];
    const float* Wp  = (const float*)d_in[18];
    const float* bp  = (const float*)d_in[19];

    const int E = in_sizes[2];

    float* out_x = (float*)d_out;                       // [NX, 64]
    float* out_c = (float*)d_out + (size_t)NX_ * DH_;   // [NC, 64]

    float* agg  = (float*)d_ws;                         // [NX, 64]
    float* h_ff = agg  + (size_t)NX_ * DH_;             // [NX, 64]
    float* h_bb = h_ff + (size_t)NX_ * DH_;             // [NX, 64]
    float* h_cx = h_bb + (size_t)NX_ * DH_;             // [NC, 64]

    // 1) node transforms for x: agg = relu(x@Wx^T+bx) (self term, also
    //    initializes agg), h_ff, h_bb.  NX % 16 == 0.
    gemm64_multi<3, true><<<512, 128, 0, stream>>>(
        x, NX_ / 16,
        Wx,  bx,  agg,
        Wff, bff, h_ff,
        Wbb, bbb, h_bb);

    // 2) node transforms for c: out_c = relu(c@Wc^T+bc), h_cx.
    gemm64_multi<2, true><<<256, 128, 0, stream>>>(
        c, NC_ / 16,
        Wc,  bc,  out_c,
        Wcx, bcx, h_cx,
        nullptr, nullptr, nullptr);

    // 3) edge scatter-adds into agg.
    int sblocks = (E * 16 + 255) / 256;
    // m1: gather e_fwd_src, scatter e_fwd_dst
    scatter_add64<<<sblocks, 256, 0, stream>>>(h_ff, e_fwd_src, e_fwd_dst, agg, E);
    // m2: gather e_bwd_dst, scatter e_bwd_src (direction swap in reference)
    scatter_add64<<<sblocks, 256, 0, stream>>>(h_bb, e_bwd_dst, e_bwd_src, agg, E);
    // m3: gather e_cx_src (context h), scatter e_cx_dst
    scatter_add64<<<sblocks, 256, 0, stream>>>(h_cx, e_cx_src, e_cx_dst, agg, E);

    // 4) pooling GEMM: out_x = agg @ Wp^T + bp (no relu).
    gemm64_multi<1, false><<<512, 128, 0, stream>>>(
        agg, NX_ / 16,
        Wp, bp, out_x,
        nullptr, nullptr, nullptr,
        nullptr, nullptr, nullptr);
}